// MNIST_QNN_Classifier_730144440591
// MI455X (gfx1250) — compile-verified
//
#include <hip/hip_runtime.h>
#include <hip/hip_bf16.h>
#include <math.h>

#define MQ    2
#define NQ    10
#define NL    8
#define NB    4
#define DIMQ  1024
#define KC    256     // K-chunk staged in LDS per TDM op
#define TN    64      // N (batch) rows per workgroup tile
#define TM    128     // M rows per workgroup tile (16 per wave x 8 waves)

typedef float v2f __attribute__((ext_vector_type(2)));
typedef float v8f __attribute__((ext_vector_type(8)));
typedef unsigned int v4u __attribute__((ext_vector_type(4)));
typedef int v4i __attribute__((ext_vector_type(4)));
typedef int v8i __attribute__((ext_vector_type(8)));

// ---------------------------------------------------------------------------
// Phase 1: materialize QT[j][k][m] = (Q_j)[m][k]  (column k of Q_j contiguous)
// One workgroup per (block j, basis column k). State lives in LDS.
// ---------------------------------------------------------------------------
__global__ __launch_bounds__(256) void qnn_build_cols(
    const float* __restrict__ thetas,   // (NB, NL*NQ)
    float* __restrict__ QT)             // (NB, DIMQ, DIMQ)
{
    __shared__ float s[DIMQ];
    const int col = blockIdx.x & (DIMQ - 1);
    const int j   = blockIdx.x >> 10;
    const int tid = threadIdx.x;

    // |e_col>
    for (int r = tid; r < DIMQ; r += 256) s[r] = (r == col) ? 1.0f : 0.0f;
    __syncthreads();

    const float* th   = thetas + j * (NL * NQ);
    const int   start = j & 1;          // entangler brick variant

    for (int l = 0; l < NL; ++l) {
        // --- per-qubit RY rotations (kron of 2x2, qubit 0 = MSB) ---
        for (int q = 0; q < NQ; ++q) {
            const float t  = 0.5f * th[l * NQ + q];
            const float c  = __cosf(t);
            const float sn = __sinf(t);
            const int   bp   = (NQ - 1) - q;   // bit position of qubit q
            const int   mask = 1 << bp;
            __syncthreads();
            #pragma unroll
            for (int pp = 0; pp < 2; ++pp) {
                const int p  = tid + pp * 256;          // pair index 0..511
                const int lo = p & (mask - 1);
                const int i0 = ((p ^ lo) << 1) | lo;    // insert 0 at bit bp
                const int i1 = i0 | mask;
                const float a0 = s[i0], a1 = s[i1];
                s[i0] = c  * a0 - sn * a1;              // RY = [[c,-s],[s,c]]
                s[i1] = sn * a0 + c  * a1;
            }
        }
        // --- CNOT brick layer: new[r] = s[perm[r]] (XOR involution) ---
        __syncthreads();
        float v[4];
        #pragma unroll
        for (int pp = 0; pp < 4; ++pp) {
            const int r = tid + pp * 256;
            int pr = r;
            for (int cq = start; cq < NQ - 1; cq += 2)
                pr ^= ((r >> ((NQ - 1) - cq)) & 1) << ((NQ - 2) - cq);
            v[pp] = s[pr];
        }
        __syncthreads();
        #pragma unroll
        for (int pp = 0; pp < 4; ++pp) s[tid + pp * 256] = v[pp];
    }
    __syncthreads();

    float* dst = QT + ((size_t)j * DIMQ + col) * DIMQ;
    for (int r = tid; r < DIMQ; r += 256) dst[r] = s[r];
}

// ---------------------------------------------------------------------------
// Phase 2: Yt[j][b][m] = sum_k Q_j[m][k] * x[b][k]
// Cooperative workgroup GEMM: 8 waves, 128M x 64N tile per workgroup.
// x chunk (TN rows x KC cols) staged global->LDS by the Tensor Data Mover
// (tensor_load_to_lds, TENSORcnt); each wave holds 4 N-subtile accumulators
// and reuses its A fragment across them (V_WMMA_F32_16X16X4_F32).
// ---------------------------------------------------------------------------
__global__ __launch_bounds__(256) void qnn_gemm_wmma(
    const float* __restrict__ QT,   // (NB, K=DIMQ, M=DIMQ)  == Q_j transposed
    const float* __restrict__ x,    // (B, DIMQ)
    float* __restrict__ Yt,         // (NB, B, DIMQ)
    int ng_cnt, int B)
{
    extern __shared__ float lds_x[];            // TN * KC floats = 64 KB

    const int wg   = blockIdx.x;
    const int ng   = wg % ng_cnt;               // N-group (64 batch rows)
    const int mg   = (wg / ng_cnt) & 7;         // M-group (128 rows of Q)
    const int j    = wg / (ng_cnt * 8);         // LCU block
    const int wave = threadIdx.x >> 5;
    const int lane = threadIdx.x & 31;
    const int half = lane >> 4;                 // K-half selector for frags
    const int lm   = lane & 15;

    const int m0     = mg * TM + wave * 16;     // this wave's 16 M rows
    const int n_base = ng * TN;                 // 64 batch rows of this WG

    const float* Qj = QT + (size_t)j * DIMQ * DIMQ;

    v8f acc0 = {}, acc1 = {}, acc2 = {}, acc3 = {};

    for (int kc = 0; kc < DIMQ; kc += KC) {
        __syncthreads();                        // previous chunk fully consumed
        if (threadIdx.x < 32) {
            // ---- Tensor DMA descriptor: 2-D tile (KC x TN) of x -> LDS ----
            const unsigned long long ga =
                (unsigned long long)(const void*)(x + (size_t)n_base * DIMQ + kc);
            const unsigned lds_base = (unsigned)(size_t)(void*)lds_x;

            v4u g0;
            g0.x = 1u;                               // count=1, user mode
            g0.y = lds_base;                         // lds_addr (bytes)
            g0.z = (unsigned)(ga & 0xffffffffu);     // global_addr[31:0]
            g0.w = (unsigned)((ga >> 32) & 0x01ffffffu) | (2u << 30); // addr[56:32] | type=2

            v8i g1;
            g1[0] = (2 << 16);                       // wg_mask=0, data_size=4B
            g1[1] = (int)((DIMQ & 0xffffu) << 16);   // tensor_dim0[15:0] = 1024
            g1[2] = (int)((DIMQ >> 16) & 0xffffu) |  // tensor_dim0[31:16]
                    (int)(((unsigned)B & 0xffffu) << 16); // tensor_dim1[15:0]
            g1[3] = (int)(((unsigned)B >> 16) & 0xffffu) |
                    (KC << 16);                      // tile_dim0 = KC (contig K)
            g1[4] = TN;                              // tile_dim1 = 64 rows, tile_dim2=0
            g1[5] = DIMQ;                            // tensor_dim0_stride[31:0] = 1024
            g1[6] = 0;                               // stride hi / dim1_stride lo
            g1[7] = 0;                               // dim1_stride hi
            v4i gz4 = {0, 0, 0, 0};
            v8i gz8 = {0, 0, 0, 0, 0, 0, 0, 0};

            __builtin_amdgcn_tensor_load_to_lds(g0, g1, gz4, gz4, gz8, 0);
            __builtin_amdgcn_s_wait_tensorcnt(0);
        }
        __syncthreads();                        // LDS chunk visible to all waves

        #pragma unroll 2
        for (int k0 = kc; k0 < kc + KC; k0 += 4) {
            const int ka = k0 + half * 2;
            v2f a;
            a.x = Qj[(size_t)ka * DIMQ + m0 + lm];           // A[m][ka]
            a.y = Qj[(size_t)(ka + 1) * DIMQ + m0 + lm];     // A[m][ka+1]

            const int kk = ka - kc;                          // 0..KC-2, even
            v2f b0 = *(const v2f*)(lds_x + (0 * 16 + lm) * KC + kk);
            v2f b1 = *(const v2f*)(lds_x + (1 * 16 + lm) * KC + kk);
            v2f b2 = *(const v2f*)(lds_x + (2 * 16 + lm) * KC + kk);
            v2f b3 = *(const v2f*)(lds_x + (3 * 16 + lm) * KC + kk);

            acc0 = __builtin_amdgcn_wmma_f32_16x16x4_f32(false, a, false, b0, (short)0, acc0, false, false);
            acc1 = __builtin_amdgcn_wmma_f32_16x16x4_f32(false, a, false, b1, (short)0, acc1, false, false);
            acc2 = __builtin_amdgcn_wmma_f32_16x16x4_f32(false, a, false, b2, (short)0, acc2, false, false);
            acc3 = __builtin_amdgcn_wmma_f32_16x16x4_f32(false, a, false, b3, (short)0, acc3, false, false);
        }
    }

    // D frag: VGPR v -> (M = v + 8*half, N = lm); store Yt[j][b][m]
    const int mst = m0 + 8 * half;
    #pragma unroll
    for (int t = 0; t < 4; ++t) {
        const v8f accv = (t == 0) ? acc0 : (t == 1) ? acc1 : (t == 2) ? acc2 : acc3;
        float* dst = Yt + (((size_t)j * B) + n_base + t * 16 + lm) * DIMQ + mst;
        #pragma unroll
        for (int v = 0; v < 8; ++v) dst[v] = accv[v];
    }
}

// ---------------------------------------------------------------------------
// Phase 3: out[b][q] = sum_j w_j * sum_k z_q[k] * Yt[j][b][k]^2
//          w_j = E[j,0]^2 = { (c0c1)^2, (c0s1)^2, (s0c2)^2, (s0s2)^2 }
// One workgroup per batch element.
// ---------------------------------------------------------------------------
__global__ __launch_bounds__(256) void qnn_expect(
    const float* __restrict__ Yt,     // (NB, B, DIMQ)
    const float* __restrict__ alpha,  // (3,)
    float* __restrict__ out,          // (B, NQ)
    int B)
{
    const int b   = blockIdx.x;
    const int tid = threadIdx.x;

    const float c0 = __cosf(0.5f * alpha[0]), s0 = __sinf(0.5f * alpha[0]);
    const float c1 = __cosf(0.5f * alpha[1]), s1 = __sinf(0.5f * alpha[1]);
    const float c2 = __cosf(0.5f * alpha[2]), s2 = __sinf(0.5f * alpha[2]);
    float w[NB] = { c0 * c1, c0 * s1, s0 * c2, s0 * s2 };
    #pragma unroll
    for (int j = 0; j < NB; ++j) w[j] *= w[j];

    float acc[NQ];
    #pragma unroll
    for (int q = 0; q < NQ; ++q) acc[q] = 0.0f;

    for (int i = tid; i < NB * DIMQ; i += 256) {
        const int j = i >> 10;
        const int k = i & (DIMQ - 1);
        const float y = Yt[((size_t)j * B + b) * DIMQ + k];
        const float p = w[j] * y * y;
        #pragma unroll
        for (int q = 0; q < NQ; ++q)
            acc[q] += ((k >> ((NQ - 1) - q)) & 1) ? -p : p;
    }

    __shared__ float red[256];
    for (int q = 0; q < NQ; ++q) {
        red[tid] = acc[q];
        __syncthreads();
        for (int step = 128; step > 0; step >>= 1) {
            if (tid < step) red[tid] += red[tid + step];
            __syncthreads();
        }
        if (tid == 0) out[b * NQ + q] = red[0];
        __syncthreads();
    }
}

// ---------------------------------------------------------------------------
extern "C" void kernel_launch(void* const* d_in, const int* in_sizes, int n_in,
                              void* d_out, int out_size, void* d_ws, size_t ws_size,
                              hipStream_t stream)
{
    const float* x      = (const float*)d_in[0];   // (B, 1024)
    const float* alpha  = (const float*)d_in[1];   // (3,)
    const float* thetas = (const float*)d_in[2];   // (4, 80)
    float* out = (float*)d_out;

    const int B      = in_sizes[0] / DIMQ;         // 256
    const int ng_cnt = B / TN;                     // 4 N-groups of 64 batches

    float* QT = (float*)d_ws;                      // NB*1024*1024 f32 = 16 MB
    float* Yt = QT + (size_t)NB * DIMQ * DIMQ;     // NB*B*1024  f32 =  4 MB

    qnn_build_cols<<<NB * DIMQ, 256, 0, stream>>>(thetas, QT);

    const size_t lds_bytes = (size_t)TN * KC * sizeof(float);  // 64 KB
    qnn_gemm_wmma<<<NB * 8 * ng_cnt, 256, lds_bytes, stream>>>(QT, x, Yt, ng_cnt, B);

    qnn_expect<<<B, 256, 0, stream>>>(Yt, alpha, out, B);
}